// BernoulliEdge_23038204575810
// MI455X (gfx1250) — compile-verified
//
#include <hip/hip_runtime.h>
#include <hip/hip_bf16.h>

#define B_DIM 32
#define N_DIM 1024
#define D_DIM 64
#define NUM_EDGES 5
#define TOTAL_ELEMS (B_DIM * N_DIM * N_DIM)          // 33554432
#define HALF_ELEMS  (TOTAL_ELEMS / 2)                // 16777216

typedef __attribute__((ext_vector_type(2))) float v2f;
typedef __attribute__((ext_vector_type(8))) float v8f;

// ---------------------------------------------------------------------------
// threefry2x32-20 (JAX-compatible rounds/key schedule)
// ---------------------------------------------------------------------------
__device__ __forceinline__ unsigned rotl32(unsigned x, int r) {
    return (x << r) | (x >> (32 - r));
}

__device__ __forceinline__ void threefry2x32(unsigned k0, unsigned k1,
                                             unsigned c0, unsigned c1,
                                             unsigned& o0, unsigned& o1) {
    const unsigned ks0 = k0, ks1 = k1, ks2 = k0 ^ k1 ^ 0x1BD11BDAu;
    unsigned x0 = c0 + ks0, x1 = c1 + ks1;
    const int r1[4] = {13, 15, 26, 6};
    const int r2[4] = {17, 29, 16, 24};
#pragma unroll
    for (int i = 0; i < 4; ++i) { x0 += x1; x1 = rotl32(x1, r1[i]); x1 ^= x0; }
    x0 += ks1; x1 += ks2 + 1u;
#pragma unroll
    for (int i = 0; i < 4; ++i) { x0 += x1; x1 = rotl32(x1, r2[i]); x1 ^= x0; }
    x0 += ks2; x1 += ks0 + 2u;
#pragma unroll
    for (int i = 0; i < 4; ++i) { x0 += x1; x1 = rotl32(x1, r1[i]); x1 ^= x0; }
    x0 += ks0; x1 += ks1 + 3u;
#pragma unroll
    for (int i = 0; i < 4; ++i) { x0 += x1; x1 = rotl32(x1, r2[i]); x1 ^= x0; }
    x0 += ks1; x1 += ks2 + 4u;
#pragma unroll
    for (int i = 0; i < 4; ++i) { x0 += x1; x1 = rotl32(x1, r1[i]); x1 ^= x0; }
    x0 += ks2; x1 += ks0 + 5u;
    o0 = x0; o1 = x1;
}

// ---------------------------------------------------------------------------
// Kernel 1: logits[b*N + r] = (tanh([curr_b ‖ nodes_br] @ W1 + b1)) @ W2 + b2
// One wave per 16-row tile; V_WMMA_F32_16X16X4_F32 chained over K=128,
// 4 accumulators covering the 64 output columns. W1 staged transposed in LDS.
// ---------------------------------------------------------------------------
#define LDS_STRIDE 132   // 128 + 4 padding floats: avoids bank conflicts

__global__ void __launch_bounds__(256)
logits_kernel(const float* __restrict__ nodes, const int* __restrict__ num_nodes,
              const float* __restrict__ W1, const float* __restrict__ b1,
              const float* __restrict__ W2, const float* __restrict__ b2,
              float* __restrict__ logits) {
    __shared__ float ldsW1T[D_DIM * LDS_STRIDE];   // W1 transposed: [n][k]

    // Stage W1 (128x64 row-major) transposed into LDS, one time per block.
    for (int idx = threadIdx.x; idx < 2 * D_DIM * D_DIM; idx += blockDim.x) {
        int kk = idx >> 6;    // 0..127
        int n  = idx & 63;    // 0..63
        ldsW1T[n * LDS_STRIDE + kk] = W1[idx];
    }
    __syncthreads();

    const int lane = threadIdx.x & 31;
    const int wave = threadIdx.x >> 5;
    const int tile = blockIdx.x * 8 + wave;        // 0..2047
    const int rbase = tile * 16;                   // flat (b*N + r) base
    const int b = rbase >> 10;
    const int nn = num_nodes[b];
    const int m = lane & 15;
    const bool hi = lane >= 16;
    const int row = (rbase & (N_DIM - 1)) + m;     // row index within batch

    const float* nodeB = nodes + (size_t)b * N_DIM * D_DIM;
    const float* currP = nodeB + (size_t)nn * D_DIM;   // curr = nodes[b, nn]
    const float* rowP  = nodeB + (size_t)row * D_DIM;

    v8f acc[4];
#pragma unroll
    for (int nt = 0; nt < 4; ++nt) {
        v8f z = {0.f, 0.f, 0.f, 0.f, 0.f, 0.f, 0.f, 0.f};
        acc[nt] = z;
    }

    const int col = lane & 15;
    // K = 128 in steps of 4 (one f32 WMMA consumes K=4)
    for (int k = 0; k < 32; ++k) {
        const int k0 = 4 * k + (hi ? 2 : 0);
        const float* src = (k0 < D_DIM) ? (currP + k0) : (rowP + (k0 - D_DIM));
        v2f a;
        a.x = src[0];
        a.y = src[1];
#pragma unroll
        for (int nt = 0; nt < 4; ++nt) {
            const int n = nt * 16 + col;
            v2f bf;
            bf.x = ldsW1T[n * LDS_STRIDE + k0];
            bf.y = ldsW1T[n * LDS_STRIDE + k0 + 1];
            acc[nt] = __builtin_amdgcn_wmma_f32_16x16x4_f32(
                false, a, false, bf, (short)0, acc[nt], false, false);
        }
    }

    // Epilogue: h = tanh(acc + b1), logits_row = sum_n h*W2[n] + b2.
    float w2c[4], b1c[4];
#pragma unroll
    for (int nt = 0; nt < 4; ++nt) {
        w2c[nt] = W2[nt * 16 + col];
        b1c[nt] = b1[nt * 16 + col];
    }
    const float bb2 = b2[0];

    float p[8];
#pragma unroll
    for (int v = 0; v < 8; ++v) {
        float s = 0.f;
#pragma unroll
        for (int nt = 0; nt < 4; ++nt)
            s += tanhf(acc[nt][v] + b1c[nt]) * w2c[nt];
        p[v] = s;
    }
    // Butterfly-sum over the 16 lanes of each half (columns of the tile).
#pragma unroll
    for (int off = 1; off < 16; off <<= 1) {
#pragma unroll
        for (int v = 0; v < 8; ++v)
            p[v] += __shfl_xor(p[v], off, 32);
    }
    if (col == 0) {
        const int mbase = hi ? 8 : 0;   // lanes 0-15 hold M=0..7, lanes 16-31 hold M=8..15
#pragma unroll
        for (int v = 0; v < 8; ++v)
            logits[rbase + mbase + v] = p[v] + bb2;
    }
}

// ---------------------------------------------------------------------------
// Kernel 2: one wave per (b, r) row. Fuses:
//   out_weights row = copy / logits-row-update
//   5x Gumbel-argmax edge sampling -> OR into adj, zero diagonal
// Streaming 4KB rows with non-temporal loads/stores (write-once 256MB output).
// ---------------------------------------------------------------------------
__global__ void __launch_bounds__(256)
adj_kernel(const float* __restrict__ adj_in, const float* __restrict__ weights_in,
           const int* __restrict__ num_nodes, const float* __restrict__ logits,
           float* __restrict__ out_adj, float* __restrict__ out_w) {
    const int lane = threadIdx.x & 31;
    const int wave = threadIdx.x >> 5;
    const int rg = blockIdx.x * 8 + wave;          // 0..32767 (b*N + r)
    const int b = rg >> 10;
    const int r = rg & (N_DIM - 1);
    const int nn = num_nodes[b];
    const size_t rowOff = (size_t)rg * N_DIM;
    const float* logB = logits + (size_t)b * N_DIM;

    // Effective weights row (post row-update), copied to output.
    float wv[32];
#pragma unroll
    for (int j = 0; j < 32; ++j) {
        const int c = j * 32 + lane;
        float x = __builtin_nontemporal_load(&weights_in[rowOff + c]);
        if (r == nn && c < nn) x = logB[c];
        wv[j] = x;
        __builtin_nontemporal_store(x, &out_w[rowOff + c]);
    }

    // Folded keys: fold_in(key(42), i) = threefry(key=(0,42), counter=(0,i))
    unsigned fk0[NUM_EDGES], fk1[NUM_EDGES];
#pragma unroll
    for (int i = 0; i < NUM_EDGES; ++i)
        threefry2x32(0u, 42u, 0u, (unsigned)i, fk0[i], fk1[i]);

    unsigned selbits = 0u;
    const float NEG = -1000000000.0f;

    for (int i = 0; i < NUM_EDGES; ++i) {
        float best = -3.402823e38f;
        int bestc = 0;
#pragma unroll
        for (int j = 0; j < 32; ++j) {
            const int c = j * 32 + lane;
            const unsigned flat = (unsigned)rowOff + (unsigned)c;
            unsigned o0, o1, bits;
            if (flat < (unsigned)HALF_ELEMS) {
                threefry2x32(fk0[i], fk1[i], flat, flat + (unsigned)HALF_ELEMS, o0, o1);
                bits = o0;
            } else {
                threefry2x32(fk0[i], fk1[i], flat - (unsigned)HALF_ELEMS, flat, o0, o1);
                bits = o1;
            }
            float u = __uint_as_float((bits >> 9) | 0x3f800000u) - 1.0f;
            u = 1e-10f + u * (1.0f - 1e-10f);
            const float g = -logf(-logf(u));                 // Gumbel(0,1)
            const float pert = (c <= nn) ? (wv[j] + g) : NEG;
            if (pert > best || (pert == best && c < bestc)) { best = pert; bestc = c; }
        }
        // Wave-wide argmax, first-index tie-break.
#pragma unroll
        for (int off = 16; off >= 1; off >>= 1) {
            const float ov = __shfl_xor(best, off, 32);
            const int oc = __shfl_xor(bestc, off, 32);
            if (ov > best || (ov == best && oc < bestc)) { best = ov; bestc = oc; }
        }
        if ((bestc & 31) == lane) selbits |= 1u << (bestc >> 5);
    }

    const bool rowvalid = (r <= nn);
#pragma unroll
    for (int j = 0; j < 32; ++j) {
        const int c = j * 32 + lane;
        const float a = __builtin_nontemporal_load(&adj_in[rowOff + c]);
        const float s = (rowvalid && ((selbits >> j) & 1u)) ? 1.0f : 0.0f;
        float outv = s + a - s * a;                 // probabilistic OR
        if (c == r) outv = 0.0f;                    // zero diagonal
        __builtin_nontemporal_store(outv, &out_adj[rowOff + c]);
    }
}

// ---------------------------------------------------------------------------
// Launch. Inputs: 0 nodes, 1 adj, 2 weights, 3 num_nodes, 4 B, 5 W1, 6 b1,
// 7 W2, 8 b2. Output: [new_adj | weights] flat f32.
// ---------------------------------------------------------------------------
extern "C" void kernel_launch(void* const* d_in, const int* in_sizes, int n_in,
                              void* d_out, int out_size, void* d_ws, size_t ws_size,
                              hipStream_t stream) {
    const float* nodes = (const float*)d_in[0];
    const float* adj   = (const float*)d_in[1];
    const float* wts   = (const float*)d_in[2];
    const int*   nn    = (const int*)d_in[3];
    const float* W1    = (const float*)d_in[5];
    const float* b1    = (const float*)d_in[6];
    const float* W2    = (const float*)d_in[7];
    const float* b2    = (const float*)d_in[8];

    float* out_adj = (float*)d_out;
    float* out_w   = out_adj + (size_t)TOTAL_ELEMS;
    float* logits  = (float*)d_ws;   // B*N floats = 128 KB scratch

    // 2048 row-tiles of 16 rows, 8 waves per 256-thread block -> 256 blocks
    logits_kernel<<<dim3(256), dim3(256), 0, stream>>>(nodes, nn, W1, b1, W2, b2, logits);
    // 32768 rows, 8 waves per block -> 4096 blocks
    adj_kernel<<<dim3(4096), dim3(256), 0, stream>>>(adj, wts, nn, logits, out_adj, out_w);
}